// CompetingRisks_36515811951154
// MI455X (gfx1250) — compile-verified
//
#include <hip/hip_runtime.h>
#include <math.h>

#define BATCH 4096
#define DDIM  128
#define RRISK 2
#define INDIM 130
#define HID   260
#define HIDP  272        // 17 tiles of 16
#define NBLK  16         // StepA blocks per risk (4096/256)
#define TPB   256
#define H0F   0.05f
#define RTOLF 1e-3f
#define ATOLF 1e-3f
#define MAXSTEPS 16

typedef __attribute__((ext_vector_type(2))) float v2f;
typedef __attribute__((ext_vector_type(8))) float v8f;

// ---------------------------------------------------------------------------
// Kernel 1: Zb_T[r][n][m] = sum_d z[m][d] * w1[r][n][2+d] + b1[r][n]
// fp32 WMMA 16x16x4, one wave per 16x16 output tile.
// A (16x4 fp32): lanes 0-15 M=0..15 {K0,K1}, lanes 16-31 M=0..15 {K2,K3}
// B (4x16 fp32): mirrored layout, N = lane&15.
// D (16x16 f32): VGPR i -> M = i + (lane<16?0:8), N = lane&15.
// Padding rows n in [260,272) of ZbT are never read by the ODE kernels, so
// out-of-range lanes just clamp to row HID-1 (no divergent masking needed).
// ---------------------------------------------------------------------------
__global__ void k_zproj_wmma(const float* __restrict__ z,
                             const float* __restrict__ w1,
                             const float* __restrict__ b1,
                             float* __restrict__ ZbT) {
  const int mt   = blockIdx.x;          // 0..255   (batch tiles)
  const int nt   = blockIdx.y;          // 0..16    (hidden tiles)
  const int r    = blockIdx.z;          // 0..1
  const int lane = threadIdx.x;         // 0..31
  const int half = lane >> 4;           // 0/1 -> K pair select
  const int l15  = lane & 15;
  const int kb   = half * 2;

  const int n   = nt * 16 + l15;        // hidden index (output row)
  const int ncl = (n < HID) ? n : (HID - 1);   // clamp: pad rows hold dup data

  const float* zrow = z + (size_t)(mt * 16 + l15) * DDIM;
  const float* wrow = w1 + ((size_t)r * HID + ncl) * INDIM + 2;

  v8f acc = {0.f, 0.f, 0.f, 0.f, 0.f, 0.f, 0.f, 0.f};
  for (int kk = 0; kk < DDIM; kk += 4) {
    // 8B-aligned: even-float offsets into fp32 arrays
    v2f a = *(const v2f*)(zrow + kk + kb);
    v2f b = *(const v2f*)(wrow + kk + kb);
    acc = __builtin_amdgcn_wmma_f32_16x16x4_f32(
        /*neg_a=*/false, a, /*neg_b=*/false, b,
        /*c_mod=*/(short)0, acc, /*reuse_a=*/false, /*reuse_b=*/false);
  }

  const float bias = b1[(size_t)r * HID + ncl];
  float* dst = ZbT + (size_t)r * HIDP * BATCH + (size_t)n * BATCH
             + (size_t)mt * 16 + half * 8;
#pragma unroll
  for (int i = 0; i < 8; ++i) dst[i] = acc[i] + bias;
}

// ---------------------------------------------------------------------------
// RHS eval: f(t1, y) for one batch element.
// zb points at ZbT row 0, column b (stride BATCH between hidden rows).
// ---------------------------------------------------------------------------
__device__ __forceinline__ float feval(const float* __restrict__ zb,
                                       const float* __restrict__ s_at,
                                       const float* __restrict__ s_ay,
                                       const float* __restrict__ s_w2,
                                       float b2v, float te, float t1s, float ys) {
  const float tt = t1s * te;
  float acc = 0.f;
#pragma unroll 4
  for (int j = 0; j < HID; ++j) {
    float pre = zb[(size_t)j * BATCH] + tt * s_at[j] + ys * s_ay[j];
    pre = fmaxf(pre, 0.f);
    acc = fmaf(pre, s_w2[j], acc);
  }
  acc += b2v;
  // numerically stable softplus
  const float sp = fmaxf(acc, 0.f) + log1pf(expf(-fabsf(acc)));
  return sp * te;
}

__device__ __forceinline__ void load_vecs(int r, const float* __restrict__ w1,
                                          const float* __restrict__ w2,
                                          float* s_at, float* s_ay, float* s_w2) {
  for (int j = threadIdx.x; j < HID; j += TPB) {
    const float* wr = w1 + ((size_t)r * HID + j) * INDIM;
    s_at[j] = wr[0];
    s_ay[j] = wr[1];
    s_w2[j] = w2[(size_t)r * HID + j];
  }
  __syncthreads();
}

// ws scalar block Q: per risk {t, h, done, accept}
// ---------------------------------------------------------------------------
__global__ void k_init(const float* __restrict__ t_eval,
                       const float* __restrict__ w1,
                       const float* __restrict__ w2,
                       const float* __restrict__ b2,
                       const float* __restrict__ ZbT,
                       float* __restrict__ Sy, float* __restrict__ Sfy,
                       float* __restrict__ Qq) {
  __shared__ float s_at[HID], s_ay[HID], s_w2[HID];
  const int r = blockIdx.y;
  const int b = blockIdx.x * TPB + threadIdx.x;
  load_vecs(r, w1, w2, s_at, s_ay, s_w2);

  const float te = t_eval[b];
  const float* zb = ZbT + (size_t)r * HIDP * BATCH + b;
  const float f0 = feval(zb, s_at, s_ay, s_w2, b2[r], te, 0.f, 0.f);

  const int ix = r * BATCH + b;
  Sy[ix]  = 0.f;
  Sfy[ix] = f0;
  if (b == 0) {
    Qq[r * 4 + 0] = 0.f;   // t
    Qq[r * 4 + 1] = H0F;   // h
    Qq[r * 4 + 2] = 0.f;   // done
    Qq[r * 4 + 3] = 0.f;   // accept (previous step)
  }
}

// ---------------------------------------------------------------------------
__global__ void k_stepA(const float* __restrict__ t_eval,
                        const float* __restrict__ w1,
                        const float* __restrict__ w2,
                        const float* __restrict__ b2,
                        const float* __restrict__ ZbT,
                        float* __restrict__ Sy, float* __restrict__ Sfy,
                        float* __restrict__ Sy5, float* __restrict__ Sk7,
                        float* __restrict__ Pp, const float* __restrict__ Qq) {
  __shared__ float s_at[HID], s_ay[HID], s_w2[HID];
  __shared__ float red[TPB];
  const int r  = blockIdx.y;
  const int b  = blockIdx.x * TPB + threadIdx.x;
  const int ix = r * BATCH + b;

  const float t     = Qq[r * 4 + 0];
  const float hstep = Qq[r * 4 + 1];
  const bool  done  = Qq[r * 4 + 2] != 0.f;
  const bool  pacc  = Qq[r * 4 + 3] != 0.f;

  // commit previous step's candidate if it was accepted
  float ys = Sy[ix];
  float f1 = Sfy[ix];
  if (pacc) {
    ys = Sy5[ix];
    f1 = Sk7[ix];
    Sy[ix]  = ys;
    Sfy[ix] = f1;
  }
  if (done) {                       // uniform per risk: whole grid-slice idles
    if (threadIdx.x == 0) Pp[r * NBLK + blockIdx.x] = 0.f;
    return;
  }

  load_vecs(r, w1, w2, s_at, s_ay, s_w2);
  const float b2v = b2[r];
  const float te  = t_eval[b];
  const float* zb = ZbT + (size_t)r * HIDP * BATCH + b;

  // warm L2/L0 for the Zb column this lane will stream 6 times
#pragma unroll
  for (int j = 0; j < HID; j += 32) __builtin_prefetch(zb + (size_t)j * BATCH, 0, 1);

  const float h = fminf(hstep, 1.0f - t);

  const float k1 = f1;
  const float k2 = feval(zb, s_at, s_ay, s_w2, b2v, te, t + 0.2f * h,
                         ys + h * (0.2f * k1));
  const float k3 = feval(zb, s_at, s_ay, s_w2, b2v, te, t + 0.3f * h,
                         ys + h * (3.f/40.f * k1 + 9.f/40.f * k2));
  const float k4 = feval(zb, s_at, s_ay, s_w2, b2v, te, t + 0.8f * h,
                         ys + h * (44.f/45.f * k1 - 56.f/15.f * k2 + 32.f/9.f * k3));
  const float k5 = feval(zb, s_at, s_ay, s_w2, b2v, te, t + 8.f/9.f * h,
                         ys + h * (19372.f/6561.f * k1 - 25360.f/2187.f * k2
                                 + 64448.f/6561.f * k3 - 212.f/729.f * k4));
  const float k6 = feval(zb, s_at, s_ay, s_w2, b2v, te, t + h,
                         ys + h * (9017.f/3168.f * k1 - 355.f/33.f * k2
                                 + 46732.f/5247.f * k3 + 49.f/176.f * k4
                                 - 5103.f/18656.f * k5));
  const float y5 = ys + h * (35.f/384.f * k1 + 500.f/1113.f * k3
                           + 125.f/192.f * k4 - 2187.f/6784.f * k5 + 11.f/84.f * k6);
  const float k7 = feval(zb, s_at, s_ay, s_w2, b2v, te, t + h, y5);

  const float err = h * (71.f/57600.f * k1 - 71.f/16695.f * k3 + 71.f/1920.f * k4
                       - 17253.f/339200.f * k5 + 22.f/525.f * k6 - 1.f/40.f * k7);
  const float scale = ATOLF + RTOLF * fmaxf(fabsf(ys), fabsf(y5));
  const float e = err / scale;

  Sy5[ix] = y5;
  Sk7[ix] = k7;

  red[threadIdx.x] = e * e;
  __syncthreads();
  for (int s = TPB / 2; s > 0; s >>= 1) {
    if (threadIdx.x < s) red[threadIdx.x] += red[threadIdx.x + s];
    __syncthreads();
  }
  if (threadIdx.x == 0) Pp[r * NBLK + blockIdx.x] = red[0];
}

// ---------------------------------------------------------------------------
__global__ void k_stepB(const float* __restrict__ Pp, float* __restrict__ Qq) {
  const int r = blockIdx.x;
  if (threadIdx.x != 0) return;

  float sum = 0.f;
  for (int i = 0; i < NBLK; ++i) sum += Pp[r * NBLK + i];

  const float t     = Qq[r * 4 + 0];
  const float hstep = Qq[r * 4 + 1];
  const bool  done  = Qq[r * 4 + 2] != 0.f;

  const float hcl = fminf(hstep, 1.0f - t);
  const float en  = sqrtf(sum / (float)BATCH);
  const bool  accept = (en <= 1.0f) && (!done);
  const float fac = fminf(fmaxf(0.9f * powf(fmaxf(en, 1e-10f), -0.2f), 0.2f), 10.0f);

  const float t_n   = accept ? (t + hcl) : t;
  const float h_n   = done ? hcl : (hcl * fac);
  const bool  doneN = done || (t_n >= 1.0f - 1e-9f);

  Qq[r * 4 + 0] = t_n;
  Qq[r * 4 + 1] = h_n;
  Qq[r * 4 + 2] = doneN ? 1.f : 0.f;
  Qq[r * 4 + 3] = accept ? 1.f : 0.f;
}

// ---------------------------------------------------------------------------
// Final: commit last candidate, haz = t_rec * f(1, yT); write haz [R,1,B]
// then chf [R,B,2,1] (= {0, yT}).
// ---------------------------------------------------------------------------
__global__ void k_final(const float* __restrict__ t_eval,
                        const float* __restrict__ w1,
                        const float* __restrict__ w2,
                        const float* __restrict__ b2,
                        const float* __restrict__ ZbT,
                        const float* __restrict__ Sy, const float* __restrict__ Sy5,
                        const float* __restrict__ Qq,
                        float* __restrict__ out) {
  __shared__ float s_at[HID], s_ay[HID], s_w2[HID];
  const int r  = blockIdx.y;
  const int b  = blockIdx.x * TPB + threadIdx.x;
  const int ix = r * BATCH + b;
  load_vecs(r, w1, w2, s_at, s_ay, s_w2);

  const bool pacc = Qq[r * 4 + 3] != 0.f;
  const float yT  = pacc ? Sy5[ix] : Sy[ix];

  const float te    = t_eval[b];
  const float t_rec = (te != 0.f) ? (1.0f / te) : 0.f;
  const float* zb   = ZbT + (size_t)r * HIDP * BATCH + b;
  const float haz   = t_rec * feval(zb, s_at, s_ay, s_w2, b2[r], te, 1.0f, yT);

  out[ix] = haz;                                       // haz [R,1,B]
  float* chf = out + RRISK * BATCH;                    // chf [R,B,2,1]
  chf[(size_t)r * BATCH * 2 + (size_t)b * 2 + 0] = 0.f;
  chf[(size_t)r * BATCH * 2 + (size_t)b * 2 + 1] = yT;
}

// ---------------------------------------------------------------------------
extern "C" void kernel_launch(void* const* d_in, const int* in_sizes, int n_in,
                              void* d_out, int out_size, void* d_ws, size_t ws_size,
                              hipStream_t stream) {
  const float* z      = (const float*)d_in[0];  // [B,D]
  const float* t_eval = (const float*)d_in[1];  // [B,1]
  const float* w1     = (const float*)d_in[2];  // [R,HID,IN]
  const float* b1     = (const float*)d_in[3];  // [R,HID]
  const float* w2     = (const float*)d_in[4];  // [R,1,HID]
  const float* b2     = (const float*)d_in[5];  // [R,1]
  float* out = (float*)d_out;

  float* ws  = (float*)d_ws;
  float* ZbT = ws;                               // [R][HIDP][B]
  float* Sy  = ws + (size_t)RRISK * HIDP * BATCH;
  float* Sfy = Sy  + RRISK * BATCH;
  float* Sy5 = Sfy + RRISK * BATCH;
  float* Sk7 = Sy5 + RRISK * BATCH;
  float* Pp  = Sk7 + RRISK * BATCH;              // [R][NBLK]
  float* Qq  = Pp  + RRISK * NBLK;               // [R][4]

  k_zproj_wmma<<<dim3(BATCH / 16, HIDP / 16, RRISK), 32, 0, stream>>>(z, w1, b1, ZbT);

  k_init<<<dim3(NBLK, RRISK), TPB, 0, stream>>>(t_eval, w1, w2, b2, ZbT, Sy, Sfy, Qq);

  for (int s = 0; s < MAXSTEPS; ++s) {
    k_stepA<<<dim3(NBLK, RRISK), TPB, 0, stream>>>(t_eval, w1, w2, b2, ZbT,
                                                   Sy, Sfy, Sy5, Sk7, Pp, Qq);
    k_stepB<<<RRISK, 32, 0, stream>>>(Pp, Qq);
  }

  k_final<<<dim3(NBLK, RRISK), TPB, 0, stream>>>(t_eval, w1, w2, b2, ZbT,
                                                 Sy, Sy5, Qq, out);
}